// GNN_encoder_10917806867253
// MI455X (gfx1250) — compile-verified
//
#include <hip/hip_runtime.h>

typedef __attribute__((ext_vector_type(2))) float v2f;
typedef __attribute__((ext_vector_type(8))) float v8f;

#ifndef BN_EPS_F
#define BN_EPS_F 1e-5f
#endif

// ---------------------------------------------------------------------------
// small utility kernels
// ---------------------------------------------------------------------------
__global__ void zero_f32_kernel(float* __restrict__ p, int n) {
  int i = blockIdx.x * blockDim.x + threadIdx.x;
  if (i < n) p[i] = 0.0f;
}

__global__ void copy_f32_kernel(float* __restrict__ dst, const float* __restrict__ src, int n) {
  int i = blockIdx.x * blockDim.x + threadIdx.x;
  if (i < n) dst[i] = src[i];
}

// ---------------------------------------------------------------------------
// edge aggregation: agg[dst[e]][d] += h[src[e]][d]  (agg pre-initialized to h
// for the GIN "x + sum" term). One float atomic per (edge, feature); 64
// consecutive lanes share an edge -> coalesced 256B gather + atomic burst.
// ---------------------------------------------------------------------------
template <int FEAT, int SHIFT>
__global__ void edge_agg_kernel(float* __restrict__ agg, const float* __restrict__ h,
                                const long long* __restrict__ srcI,
                                const long long* __restrict__ dstI,
                                long long total) {
  long long i = (long long)blockIdx.x * blockDim.x + threadIdx.x;
  if (i >= total) return;
  long long e = i >> SHIFT;
  int d = (int)(i & (FEAT - 1));
  long long s = srcI[e];
  long long t = dstI[e];
  unsafeAtomicAdd(&agg[t * FEAT + d], h[s * FEAT + d]);
}

// ---------------------------------------------------------------------------
// layer-0 first linear: (N x 2) @ (2 x 64) + b, ReLU.  K=2 -> plain VALU.
// ---------------------------------------------------------------------------
__global__ void lin0_kernel(float* __restrict__ out, const float* __restrict__ x2,
                            const float* __restrict__ W, const float* __restrict__ b,
                            int n) {
  int i = blockIdx.x * blockDim.x + threadIdx.x;
  if (i >= n * 64) return;
  int r = i >> 6, c = i & 63;
  float v = fmaf(x2[2 * r], W[c], fmaf(x2[2 * r + 1], W[64 + c], b[c]));
  out[i] = fmaxf(v, 0.0f);
}

// ---------------------------------------------------------------------------
// Y = X @ W + bias (optional ReLU).  X: n x 64, W: 64 x 64 row-major.
// Block = 128 threads = 4 waves; block covers 64 rows, each wave a 16-row
// strip. Wave does 4 col-tiles x 16 K-steps of V_WMMA_F32_16X16X4_F32.
// X tile + W staged in LDS (32 KB). Wave-uniform guard keeps EXEC all-1s.
//
// Fragment layouts (ISA 7.12.2, wave32):
//   A 16x4 f32 : lane m = lane&15 is the row, half-wave picks K pair:
//                a.x = A[m][kh], a.y = A[m][kh+1], kh = (lane>>4)*2
//   B 4x16 f32 : mirrored: col = lane&15, b.x = B[kh][col], b.y = B[kh+1][col]
//   C/D 16x16  : VGPR i -> row i (lanes 0-15) / i+8 (lanes 16-31), col = lane&15
// ---------------------------------------------------------------------------
__launch_bounds__(128)
__global__ void gemm64_wmma_kernel(float* __restrict__ Y, const float* __restrict__ X,
                                   const float* __restrict__ W, const float* __restrict__ bias,
                                   int n, int relu) {
  __shared__ float ldsW[64 * 64];
  __shared__ float ldsX[64 * 64];

  const int t = threadIdx.x;
  const int blockRow0 = blockIdx.x * 64;

  // stage W (4096 floats) as float4
  {
    const float4* Wv = (const float4*)W;
    float4* Lv = (float4*)ldsW;
    #pragma unroll
    for (int i = t; i < 1024; i += 128) Lv[i] = Wv[i];
  }
  // stage X tile (64 rows), zero-pad rows >= n
  {
    const float4* Xv = (const float4*)X;
    float4* Lx = (float4*)ldsX;
    #pragma unroll
    for (int i = t; i < 1024; i += 128) {
      int row = i >> 4;  // 16 float4 per row
      int gr = blockRow0 + row;
      float4 v = {0.0f, 0.0f, 0.0f, 0.0f};
      if (gr < n) v = Xv[(size_t)gr * 16 + (i & 15)];
      Lx[i] = v;
    }
  }
  __syncthreads();

  const int wave = t >> 5;
  const int lane = t & 31;
  const int r0 = blockRow0 + wave * 16;
  if (r0 >= n) return;  // wave-uniform: EXEC stays all-1s for WMMA

  const int m = lane & 15;
  const int kh = (lane >> 4) << 1;  // 0 or 2

  // preload all 16 A fragments for this wave's 16-row strip
  v2f afrag[16];
  {
    const float* xrow = &ldsX[(wave * 16 + m) * 64 + kh];
    #pragma unroll
    for (int kk = 0; kk < 16; ++kk) {
      v2f a;
      a.x = xrow[kk * 4];
      a.y = xrow[kk * 4 + 1];
      afrag[kk] = a;
    }
  }

  #pragma unroll
  for (int nt = 0; nt < 4; ++nt) {
    const int col = nt * 16 + m;
    v8f c = {};
    #pragma unroll
    for (int kk = 0; kk < 16; ++kk) {
      v2f b;
      b.x = ldsW[(kk * 4 + kh) * 64 + col];
      b.y = ldsW[(kk * 4 + kh + 1) * 64 + col];
      c = __builtin_amdgcn_wmma_f32_16x16x4_f32(
          /*neg_a=*/false, afrag[kk], /*neg_b=*/false, b,
          /*c_mod=*/(short)0, c, /*reuse_a=*/false, /*reuse_b=*/false);
    }
    const float bc = bias[col];
    const int rbase = r0 + ((lane >> 4) << 3);  // +8 for upper half-wave
    #pragma unroll
    for (int i = 0; i < 8; ++i) {
      int row = rbase + i;
      if (row < n) {
        float v = c[i] + bc;
        if (relu) v = fmaxf(v, 0.0f);
        Y[(size_t)row * 64 + col] = v;
      }
    }
  }
}

// ---------------------------------------------------------------------------
// BatchNorm batch statistics: per-column sum and sum-of-squares into
// stats[0..63] / stats[64..127] via LDS partials + global float atomics.
// ---------------------------------------------------------------------------
__global__ void bn_stats_kernel(const float* __restrict__ h, float* __restrict__ stats, int n) {
  __shared__ float ssum[256];
  __shared__ float ssq[256];
  const int t = threadIdx.x;
  const int col = t & 63;
  const int ro = t >> 6;  // 0..3
  float s = 0.0f, q = 0.0f;
  for (int r = blockIdx.x * 4 + ro; r < n; r += gridDim.x * 4) {
    float v = h[(size_t)r * 64 + col];
    s += v;
    q = fmaf(v, v, q);
  }
  ssum[t] = s;
  ssq[t] = q;
  __syncthreads();
  if (t < 64) {
    s = ssum[t] + ssum[t + 64] + ssum[t + 128] + ssum[t + 192];
    q = ssq[t] + ssq[t + 64] + ssq[t + 128] + ssq[t + 192];
    unsafeAtomicAdd(&stats[t], s);
    unsafeAtomicAdd(&stats[64 + t], q);
  }
}

__global__ void bn_apply_kernel(float* __restrict__ out, const float* __restrict__ h,
                                const float* __restrict__ stats,
                                const float* __restrict__ gamma, const float* __restrict__ beta,
                                int n, int relu) {
  int i = blockIdx.x * blockDim.x + threadIdx.x;
  if (i >= n * 64) return;
  int c = i & 63;
  float inv_n = 1.0f / (float)n;
  float mu = stats[c] * inv_n;
  float var = fmaf(-mu, mu, stats[64 + c] * inv_n);
  float scale = rsqrtf(var + BN_EPS_F) * gamma[c];
  float shift = fmaf(-mu, scale, beta[c]);
  float v = fmaf(h[i], scale, shift);
  if (relu) v = fmaxf(v, 0.0f);
  out[i] = v;
}

// ---------------------------------------------------------------------------
// orchestration
// ---------------------------------------------------------------------------
extern "C" void kernel_launch(void* const* d_in, const int* in_sizes, int n_in,
                              void* d_out, int out_size, void* d_ws, size_t ws_size,
                              hipStream_t stream) {
  const float* nodes      = (const float*)d_in[0];
  const long long* edges  = (const long long*)d_in[1];
  // d_in[2] graph_indicators: unused by the reference output
  const float* W1_0 = (const float*)d_in[3];
  const float* b1_0 = (const float*)d_in[4];
  const float* W2_0 = (const float*)d_in[5];
  const float* b2_0 = (const float*)d_in[6];
  const float* W1   = (const float*)d_in[7];   // (4,64,64)
  const float* b1   = (const float*)d_in[8];   // (4,64)
  const float* W2   = (const float*)d_in[9];   // (4,64,64)
  const float* b2   = (const float*)d_in[10];  // (4,64)
  const float* gamma = (const float*)d_in[11]; // (5,64)
  const float* beta  = (const float*)d_in[12]; // (5,64)

  const int N = in_sizes[0] / 2;          // D_IN == 2
  const long long E = in_sizes[1] / 2;    // edge_indexs is (2,E)
  const long long* srcIdx = edges;
  const long long* dstIdx = edges + E;

  float* bufA  = (float*)d_ws;                 // N*64
  float* bufB  = bufA + (size_t)N * 64;        // N*64
  float* agg2  = bufB + (size_t)N * 64;        // N*2 (layer-0 aggregation)
  float* stats = agg2 + (size_t)N * 2;         // 5*128
  float* outp  = (float*)d_out;

  const int TB = 256;
  const int nBlkNE = (int)(((size_t)N * 64 + TB - 1) / TB);
  const int nBlkGemm = (N + 63) / 64;

  // zero BN stats for all 5 layers (d_ws is poisoned by the harness)
  zero_f32_kernel<<<(5 * 128 + TB - 1) / TB, TB, 0, stream>>>(stats, 5 * 128);

  // -------- layer 0 --------
  copy_f32_kernel<<<((size_t)N * 2 + TB - 1) / TB, TB, 0, stream>>>(agg2, nodes, N * 2);
  {
    long long total = E * 2;
    int blocks = (int)((total + TB - 1) / TB);
    edge_agg_kernel<2, 1><<<blocks, TB, 0, stream>>>(agg2, nodes, srcIdx, dstIdx, total);
  }
  lin0_kernel<<<nBlkNE, TB, 0, stream>>>(bufA, agg2, W1_0, b1_0, N);
  gemm64_wmma_kernel<<<nBlkGemm, 128, 0, stream>>>(bufB, bufA, W2_0, b2_0, N, /*relu=*/0);
  bn_stats_kernel<<<240, 256, 0, stream>>>(bufB, stats, N);
  bn_apply_kernel<<<nBlkNE, TB, 0, stream>>>(bufA, bufB, stats, gamma, beta, N, /*relu=*/1);

  // -------- layers 1..4 --------
  for (int l = 0; l < 4; ++l) {
    const int lay = l + 1;
    // agg = h (GIN: x + sum of neighbors)
    copy_f32_kernel<<<nBlkNE, TB, 0, stream>>>(bufB, bufA, N * 64);
    {
      long long total = E * 64;
      int blocks = (int)((total + TB - 1) / TB);
      edge_agg_kernel<64, 6><<<blocks, TB, 0, stream>>>(bufB, bufA, srcIdx, dstIdx, total);
    }
    // h1 = relu(agg @ W1[l] + b1[l])
    gemm64_wmma_kernel<<<nBlkGemm, 128, 0, stream>>>(
        bufA, bufB, W1 + (size_t)l * 64 * 64, b1 + (size_t)l * 64, N, /*relu=*/1);
    // h2 = h1 @ W2[l] + b2[l]
    gemm64_wmma_kernel<<<nBlkGemm, 128, 0, stream>>>(
        bufB, bufA, W2 + (size_t)l * 64 * 64, b2 + (size_t)l * 64, N, /*relu=*/0);
    // BN (+ReLU except last layer); last layer writes d_out directly
    bn_stats_kernel<<<240, 256, 0, stream>>>(bufB, stats + (size_t)lay * 128, N);
    float* dst = (lay == 4) ? outp : bufA;
    bn_apply_kernel<<<nBlkNE, TB, 0, stream>>>(
        dst, bufB, stats + (size_t)lay * 128, gamma + (size_t)lay * 64,
        beta + (size_t)lay * 64, N, (lay < 4) ? 1 : 0);
  }
}